// GCNNet_43516608643384
// MI455X (gfx1250) — compile-verified
//
#include <hip/hip_runtime.h>

// ---------------------------------------------------------------------------
// GCN: h1 = tanh(A @ (X @ W1) + b1); h2 = tanh(A @ (h1 @ W2) + b2);
//      out = h2 @ fc_w + fc_b
// N=50000, E=600000, all dims 128, fp32.
// ---------------------------------------------------------------------------

#define N_NODES 50000
#define N_EDGES 600000
#define DIM     128
#define M_TILES (N_NODES / 16)   // 3125, exact

typedef __attribute__((ext_vector_type(2))) float v2f;
typedef __attribute__((ext_vector_type(8))) float v8f;

// ---------------------------------------------------------------------------
// Dense GEMM  C[M,128] = A[M,128] @ W[128,128]  via V_WMMA_F32_16X16X4_F32.
// Block = 128 threads = 4 wave32; wave w computes one 16-row M tile x all 128
// cols (8 accumulators of v8f). W is staged into LDS pre-swizzled into WMMA
// B-fragment order: pair index p = ((kk*8 + nt)*32 + lane), floats lds[2p..2p+1]
// = W[kk*4 + (lane>=16?2:0) + {0,1}][nt*16 + lane%16]  -> one ds_load_b64/lane.
// ---------------------------------------------------------------------------
__global__ __launch_bounds__(128)
void gcn_gemm128(const float* __restrict__ A, const float* __restrict__ W,
                 float* __restrict__ C, int M) {
    __shared__ float ldsW[DIM * DIM];   // 64 KB (<< 320 KB/WGP)

    const int tid  = threadIdx.x;
    const int lane = tid & 31;
    const int wave = tid >> 5;

    // ---- stage W into fragment order (all 128 threads cooperate) ----
    for (int p = tid; p < 8192; p += 128) {
        const int l    = p & 31;
        const int nt   = (p >> 5) & 7;
        const int kk   = p >> 8;                       // 0..31
        const int krow = kk * 4 + ((l & 16) ? 2 : 0);  // K base for this lane half
        const int col  = nt * 16 + (l & 15);
        ldsW[2 * p + 0] = W[(krow + 0) * DIM + col];
        ldsW[2 * p + 1] = W[(krow + 1) * DIM + col];
    }
    __syncthreads();

    const int mtile = blockIdx.x * 4 + wave;
    if (mtile * 16 >= M) return;                       // tail waves (after barrier)

    // A-fragment base: lane half selects K offset 0 or 2 (ISA 16x4 A layout)
    const float* __restrict__ Arow =
        A + (size_t)(mtile * 16 + (lane & 15)) * DIM + ((lane & 16) ? 2 : 0);

    v8f acc[8] = {};

    for (int kk = 0; kk < 32; ++kk) {
        const v2f a = *(const v2f*)(Arow + kk * 4);    // A[m][k], A[m][k+1]
#pragma unroll
        for (int nt = 0; nt < 8; ++nt) {
            const v2f b =
                *(const v2f*)&ldsW[(((kk << 3) + nt) * 32 + lane) * 2];
            acc[nt] = __builtin_amdgcn_wmma_f32_16x16x4_f32(
                /*neg_a=*/false, a, /*neg_b=*/false, b,
                /*c_mod=*/(short)0, acc[nt],
                /*reuse_a=*/false, /*reuse_b=*/false);
        }
    }

    // ---- store C per 16x16 f32 D layout: VGPR j -> row j (+8 for hi half) ----
    const int row0 = mtile * 16 + ((lane & 16) ? 8 : 0);
    const int colc = lane & 15;
#pragma unroll
    for (int nt = 0; nt < 8; ++nt) {
#pragma unroll
        for (int j = 0; j < 8; ++j) {
            C[(size_t)(row0 + j) * DIM + nt * 16 + colc] = acc[nt][j];
        }
    }
}

// ---------------------------------------------------------------------------
// Zero-fill (graph-capture safe; no hipMemset)
// ---------------------------------------------------------------------------
__global__ __launch_bounds__(256)
void gcn_fill0(float* __restrict__ p, int n) {
    const int i = blockIdx.x * 256 + threadIdx.x;
    if (i < n) p[i] = 0.0f;
}

// ---------------------------------------------------------------------------
// SpMM scatter: out[row[e],:] += vals[e] * h[col[e],:]
// One edge per wave32; lane handles float4 lane of the 128-float row.
// Gathers hit L2 (25.6 MB activation matrix is L2-resident); scatter via
// native fp32 global atomics (agent scope, relaxed).
// ---------------------------------------------------------------------------
__global__ __launch_bounds__(256)
void gcn_spmm(const int* __restrict__ erow, const int* __restrict__ ecol,
              const float* __restrict__ evals, const float* __restrict__ h,
              float* __restrict__ out, int E) {
    const int e    = blockIdx.x * 8 + (threadIdx.x >> 5);
    const int lane = threadIdx.x & 31;
    if (e >= E) return;

    const int   r = erow[e];
    const int   c = ecol[e];
    const float v = evals[e];

    const float4 x = ((const float4*)(h + (size_t)c * DIM))[lane];
    float* __restrict__ o = out + (size_t)r * DIM + lane * 4;

    __hip_atomic_fetch_add(o + 0, v * x.x, __ATOMIC_RELAXED, __HIP_MEMORY_SCOPE_AGENT);
    __hip_atomic_fetch_add(o + 1, v * x.y, __ATOMIC_RELAXED, __HIP_MEMORY_SCOPE_AGENT);
    __hip_atomic_fetch_add(o + 2, v * x.z, __ATOMIC_RELAXED, __HIP_MEMORY_SCOPE_AGENT);
    __hip_atomic_fetch_add(o + 3, v * x.w, __ATOMIC_RELAXED, __HIP_MEMORY_SCOPE_AGENT);
}

// ---------------------------------------------------------------------------
// In-place bias + tanh over [N, 128]
// ---------------------------------------------------------------------------
__global__ __launch_bounds__(256)
void gcn_bias_tanh(float* __restrict__ x, const float* __restrict__ b, int n) {
    const int i = blockIdx.x * 256 + threadIdx.x;
    if (i < n) x[i] = tanhf(x[i] + b[i & (DIM - 1)]);
}

// ---------------------------------------------------------------------------
// Final GEMV: out[i] = dot(h[i,:], fc_w) + fc_b ; one node per wave32
// ---------------------------------------------------------------------------
__global__ __launch_bounds__(256)
void gcn_gemv(const float* __restrict__ h, const float* __restrict__ fcw,
              const float* __restrict__ fcb, float* __restrict__ out, int M) {
    const int node = blockIdx.x * 8 + (threadIdx.x >> 5);
    const int lane = threadIdx.x & 31;
    if (node >= M) return;

    const float4 x = ((const float4*)(h + (size_t)node * DIM))[lane];
    const float4 w = ((const float4*)fcw)[lane];
    float s = x.x * w.x + x.y * w.y + x.z * w.z + x.w * w.w;
#pragma unroll
    for (int off = 16; off > 0; off >>= 1) s += __shfl_down(s, off, 32);
    if (lane == 0) out[node] = s + fcb[0];
}

// ---------------------------------------------------------------------------
extern "C" void kernel_launch(void* const* d_in, const int* in_sizes, int n_in,
                              void* d_out, int out_size, void* d_ws, size_t ws_size,
                              hipStream_t stream) {
    const float* features = (const float*)d_in[0];
    const int*   erow     = (const int*)  d_in[1];
    const int*   ecol     = (const int*)  d_in[2];
    const float* evals    = (const float*)d_in[3];
    const float* W1       = (const float*)d_in[4];
    const float* b1       = (const float*)d_in[5];
    const float* W2       = (const float*)d_in[6];
    const float* b2       = (const float*)d_in[7];
    const float* fcw      = (const float*)d_in[8];
    const float* fcb      = (const float*)d_in[9];
    float*       out      = (float*)d_out;

    float* tmp0 = (float*)d_ws;                          // [N,128] GEMM output
    float* tmp1 = tmp0 + (size_t)N_NODES * DIM;          // [N,128] SpMM output

    const int nElem      = N_NODES * DIM;
    const dim3 gemmGrid((M_TILES + 3) / 4);              // 4 m-tiles (waves)/block
    const dim3 elemGrid((nElem + 255) / 256);
    const dim3 spmmGrid((N_EDGES + 7) / 8);              // 8 edges (waves)/block
    const dim3 gemvGrid((N_NODES + 7) / 8);

    // ---- layer 1 ----
    gcn_gemm128<<<gemmGrid, 128, 0, stream>>>(features, W1, tmp0, N_NODES);
    gcn_fill0  <<<elemGrid, 256, 0, stream>>>(tmp1, nElem);
    gcn_spmm   <<<spmmGrid, 256, 0, stream>>>(erow, ecol, evals, tmp0, tmp1, N_EDGES);
    gcn_bias_tanh<<<elemGrid, 256, 0, stream>>>(tmp1, b1, nElem);

    // ---- layer 2 ----
    gcn_gemm128<<<gemmGrid, 128, 0, stream>>>(tmp1, W2, tmp0, N_NODES);
    gcn_fill0  <<<elemGrid, 256, 0, stream>>>(tmp1, nElem);
    gcn_spmm   <<<spmmGrid, 256, 0, stream>>>(erow, ecol, evals, tmp0, tmp1, N_EDGES);
    gcn_bias_tanh<<<elemGrid, 256, 0, stream>>>(tmp1, b2, nElem);

    // ---- readout ----
    gcn_gemv<<<gemvGrid, 256, 0, stream>>>(tmp1, fcw, fcb, out, N_NODES);
}